// E8RHTFusedExperts_56547539419789
// MI455X (gfx1250) — compile-verified
//
#include <hip/hip_runtime.h>
#include <hip/hip_bf16.h>

// MoE fused experts: T=16 tokens, K=8, E=64 experts, D=2048, F=1024.
// Weight-bandwidth-bound (~1.07 GB fp32 weights, ~2 FLOP/byte).
// fp32 WMMA (v_wmma_f32_16x16x4_f32) + async global->LDS double-buffered
// staging (global_load_async_to_lds_*, ASYNCcnt) + deterministic reduction.

#define MOE_T 16
#define MOE_K 8
#define MOE_E 64
#define MOE_D 2048
#define MOE_F 1024

#define KC 32          // K-chunk staged in LDS per pipeline stage
#define TILE_N 128     // N columns per block (8 wave32s x 16)
#define XS_STRIDE 36   // padded A-tile row stride (floats), 16B-aligned rows
#define WS_STRIDE 136  // padded B-tile row stride (floats), 16B-aligned rows

typedef __attribute__((ext_vector_type(2))) float v2f;
typedef __attribute__((ext_vector_type(8))) float v8f;

// ---------------------------------------------------------------------------
// helpers: CDNA5 async global->LDS copies (ASYNCcnt-tracked, bypass VGPRs)
// ---------------------------------------------------------------------------
template <typename T>
__device__ __forceinline__ uint32_t lds_u32(T* p) {
    // generic pointers into the LDS aperture keep the wave-relative LDS byte
    // address in addr[31:0] (CDNA5 ISA 10.2): truncate.
    return (uint32_t)(uintptr_t)p;
}

__device__ __forceinline__ void async_copy_b128(uint32_t lds, const float* g) {
    asm volatile("global_load_async_to_lds_b128 %0, %1, off"
                 :: "v"(lds), "v"(g) : "memory");
}
__device__ __forceinline__ void async_copy_b64(uint32_t lds, const float* g) {
    asm volatile("global_load_async_to_lds_b64 %0, %1, off"
                 :: "v"(lds), "v"(g) : "memory");
}
#define WAIT_ASYNC_LE_5() asm volatile("s_wait_asynccnt 0x5" ::: "memory")
#define WAIT_ASYNC_LE_0() asm volatile("s_wait_asynccnt 0x0" ::: "memory")

// ---------------------------------------------------------------------------
// route[t*E + e] = sum_k top_k_weights[t,k] * (top_k_index[t,k] == e)
// ---------------------------------------------------------------------------
__global__ __launch_bounds__(256) void moe_route_kernel(
    const int* __restrict__ topk_idx,    // [T,K]
    const float* __restrict__ topk_w,    // [T,K]
    float* __restrict__ route)           // [T,E]
{
    int i = threadIdx.x + blockIdx.x * 256;
    if (i < MOE_T * MOE_E) {
        int t = i >> 6;
        int e = i & (MOE_E - 1);
        float s = 0.0f;
#pragma unroll
        for (int k = 0; k < MOE_K; ++k) {
            s += (topk_idx[t * MOE_K + k] == e) ? topk_w[t * MOE_K + k] : 0.0f;
        }
        route[i] = s;
    }
}

// ---------------------------------------------------------------------------
// Unified 16xN GEMM over one expert, async double-buffered staging.
//   UP  (KDIM=D, NDIM=F): Out[e,t,n] = route[t,e] * relu(X[t,:] . Wup[e,:,n])^2
//   DOWN(KDIM=F, NDIM=D): Out[e,t,n] = A[e,t,:] . Wdown[e,:,n]
// grid = (E, NDIM/TILE_N), block = 256 (8 wave32s, one 16x16 C-tile per wave)
// ---------------------------------------------------------------------------
template <int KDIM, int NDIM, bool UP>
__global__ __launch_bounds__(256) void moe_gemm_kernel(
    const float* __restrict__ A0,     // UP: X [16,KDIM]; DOWN: Hact [E,16,KDIM]
    const float* __restrict__ Bmat,   // [E, KDIM, NDIM]
    const float* __restrict__ route,  // [T,E] (UP only)
    float* __restrict__ Out)          // [E, 16, NDIM]
{
    __shared__ float Asm0[MOE_T * XS_STRIDE];
    __shared__ float Asm1[MOE_T * XS_STRIDE];
    __shared__ float Bsm0[KC * WS_STRIDE];
    __shared__ float Bsm1[KC * WS_STRIDE];
    __shared__ float rs[MOE_T];

    const int e     = blockIdx.x;
    const int nbase = blockIdx.y * TILE_N;
    const int tid   = threadIdx.x;
    const int lane  = tid & 31;
    const int wave  = tid >> 5;
    const int half  = lane >> 4;   // 0: lanes 0-15, 1: lanes 16-31
    const int l15   = lane & 15;

    const float* __restrict__ Ap = UP ? A0 : (A0 + (size_t)e * MOE_T * KDIM);
    const float* __restrict__ Bp = Bmat + (size_t)e * KDIM * NDIM;

    if (UP && tid < MOE_T) rs[tid] = route[tid * MOE_E + e];

    // ---- per-thread staging coordinates (all 256 threads, no branches) ----
    // A tile: 16 rows x 32 floats = 256 b64 slots, 1 per thread
    const int arow = tid >> 4;
    const int ac2  = (tid & 15) << 1;
    const float* ga = Ap + (size_t)arow * KDIM + ac2;
    const uint32_t la0 = lds_u32(&Asm0[arow * XS_STRIDE + ac2]);
    const uint32_t la1 = lds_u32(&Asm1[arow * XS_STRIDE + ac2]);

    // B tile: 32 rows x 128 floats = 1024 b128 slots, 4 per thread
    const float* gb0;
    const float* gb1;
    const float* gb2;
    const float* gb3;
    uint32_t lb0[4], lb1[4];
    {
        const float* gtmp[4];
#pragma unroll
        for (int i = 0; i < 4; ++i) {
            const int idx  = tid + i * 256;
            const int brow = idx >> 5;            // 32 b128 per row
            const int bc4  = (idx & 31) << 2;
            gtmp[i] = Bp + (size_t)brow * NDIM + nbase + bc4;
            lb0[i]  = lds_u32(&Bsm0[brow * WS_STRIDE + bc4]);
            lb1[i]  = lds_u32(&Bsm1[brow * WS_STRIDE + bc4]);
        }
        gb0 = gtmp[0]; gb1 = gtmp[1]; gb2 = gtmp[2]; gb3 = gtmp[3];
    }
    const size_t bstep = (size_t)KC * NDIM;

#define STAGE(LA, LB)                                                         \
    do {                                                                      \
        async_copy_b64((LA), ga);                                             \
        async_copy_b128((LB)[0], gb0);                                        \
        async_copy_b128((LB)[1], gb1);                                        \
        async_copy_b128((LB)[2], gb2);                                        \
        async_copy_b128((LB)[3], gb3);                                        \
        __builtin_prefetch(gb0 + 3 * bstep, 0, 0); /* speculative, OOB-safe */\
        ga += KC;                                                             \
        gb0 += bstep; gb1 += bstep; gb2 += bstep; gb3 += bstep;               \
    } while (0)

    v8f acc = {};
    const int ncol_local = wave * 16 + l15;

#define COMPUTE(AS, BS)                                                       \
    do {                                                                      \
        _Pragma("unroll")                                                     \
        for (int kk = 0; kk < KC; kk += 4) {                                  \
            const int kA = kk + 2 * half;                                     \
            v2f a;                                                            \
            a.x = (AS)[l15 * XS_STRIDE + kA];                                 \
            a.y = (AS)[l15 * XS_STRIDE + kA + 1];                             \
            v2f b;                                                            \
            b.x = (BS)[(kA)     * WS_STRIDE + ncol_local];                    \
            b.y = (BS)[(kA + 1) * WS_STRIDE + ncol_local];                    \
            acc = __builtin_amdgcn_wmma_f32_16x16x4_f32(                      \
                false, a, false, b, (short)0, acc, false, false);             \
        }                                                                     \
    } while (0)

    // ---- software pipeline: chunk count KDIM/KC is even for both passes ----
    STAGE(la0, lb0);                       // chunk 0 -> buffer 0

    const int nhalf = KDIM / (2 * KC);
#pragma unroll 1
    for (int j = 0; j < nhalf; ++j) {
        // phase A: compute buf0, stage into buf1 (chunk 2j+1 always exists)
        STAGE(la1, lb1);
        WAIT_ASYNC_LE_5();                 // loads complete in order
        __syncthreads();
        COMPUTE(Asm0, Bsm0);
        __syncthreads();

        // phase B: compute buf1, stage chunk 2j+2 into buf0 if it exists
        if (j < nhalf - 1) {
            STAGE(la0, lb0);
            WAIT_ASYNC_LE_5();
        } else {
            WAIT_ASYNC_LE_0();
        }
        __syncthreads();
        COMPUTE(Asm1, Bsm1);
        __syncthreads();
    }
#undef STAGE
#undef COMPUTE

    // ---- epilogue: C-tile scatter; t = r + 8*half, col = ncol_local ----
    const int ncol = nbase + ncol_local;
#pragma unroll
    for (int r = 0; r < 8; ++r) {
        const int t = r + 8 * half;
        float v = acc[r];
        if (UP) {
            v = fmaxf(v, 0.0f);
            v = v * v * rs[t];             // fold routing weight here
        }
        Out[((size_t)e * MOE_T + t) * NDIM + ncol] = v;
    }
}

// ---------------------------------------------------------------------------
// Fixed-order (deterministic) reduction over experts: out[t,d] = sum_e P[e,t,d]
// ---------------------------------------------------------------------------
__global__ __launch_bounds__(256) void moe_reduce_kernel(
    const float* __restrict__ P,   // [E,T,D]
    float* __restrict__ out)       // [T,D]
{
    const int i = threadIdx.x + blockIdx.x * 256;
    float s = 0.0f;
#pragma unroll 8
    for (int e = 0; e < MOE_E; ++e) {
        s += P[(size_t)e * (MOE_T * MOE_D) + i];
    }
    out[i] = s;
}

// ---------------------------------------------------------------------------
extern "C" void kernel_launch(void* const* d_in, const int* in_sizes, int n_in,
                              void* d_out, int out_size, void* d_ws, size_t ws_size,
                              hipStream_t stream) {
    const float* X      = (const float*)d_in[0];  // hidden_states [T,D]
    const int*   tk_idx = (const int*)d_in[1];    // top_k_index   [T,K]
    const float* tk_w   = (const float*)d_in[2];  // top_k_weights [T,K]
    const float* Wup    = (const float*)d_in[3];  // [E,D,F]
    const float* Wdown  = (const float*)d_in[4];  // [E,F,D]
    float* out = (float*)d_out;                   // [T,D]

    // workspace layout
    char* ws = (char*)d_ws;
    float* route = (float*)ws;                                   // 4 KB
    float* Hact  = (float*)(ws + 4096);                          // 4 MB (route-scaled relu^2)
    float* P     = (float*)(ws + 4096 + (size_t)MOE_E * MOE_T * MOE_F * 4); // 8 MB

    // 1) routing matrix
    moe_route_kernel<<<dim3((MOE_T * MOE_E + 255) / 256), dim3(256), 0, stream>>>(
        tk_idx, tk_w, route);

    // 2) up projection + route * relu^2 : grid (64, 8)
    moe_gemm_kernel<MOE_D, MOE_F, true>
        <<<dim3(MOE_E, MOE_F / TILE_N), dim3(256), 0, stream>>>(X, Wup, route, Hact);

    // 3) down projection (pure GEMM) : grid (64, 16)
    moe_gemm_kernel<MOE_F, MOE_D, false>
        <<<dim3(MOE_E, MOE_D / TILE_N), dim3(256), 0, stream>>>(Hact, Wdown, nullptr, P);

    // 4) deterministic expert reduction
    moe_reduce_kernel<<<dim3((MOE_T * MOE_D) / 256), dim3(256), 0, stream>>>(P, out);
}